// EdgeReprAggregator_35639638622224
// MI455X (gfx1250) — compile-verified
//
#include <hip/hip_runtime.h>

typedef __attribute__((ext_vector_type(2))) float v2f;
typedef __attribute__((ext_vector_type(8))) float v8f;

#define KE 32              // edges per node (K)
#define EDIM 128           // E
#define FDIM 512           // 4E input features
#define THREADS 256        // 8 waves of 32
#define XSTRIDE 512        // LDS row stride in floats (xor-swizzled banks)
#define SMEM_ALPHA (KE * XSTRIDE)       // 16384 floats
#define SMEM_FLOATS (SMEM_ALPHA + KE)   // + alpha[32]

__global__ __launch_bounds__(THREADS)
void edge_agg_kernel(const float* __restrict__ msg,
                     const float* __restrict__ e_repr,
                     const float* __restrict__ alpha,
                     const float* __restrict__ curr_emb,
                     const float* __restrict__ W1,
                     const float* __restrict__ b1,
                     const float* __restrict__ W2,
                     const float* __restrict__ b2,
                     float* __restrict__ out)
{
    extern __shared__ float smem[];
    const int n    = blockIdx.x;
    const int tid  = threadIdx.x;
    const int lane = tid & 31;
    const int wave = tid >> 5;          // 0..7 -> 16-column tile
    const int half = lane >> 4;         // 0 or 1 (WMMA lane half)
    const int l15  = lane & 15;

    // ---- alpha(n, 0:32) into LDS ----
    if (tid < KE) smem[SMEM_ALPHA + tid] = alpha[(size_t)n * KE + tid];

    // ---- stage x = [msg | e_repr] (32 x 512 f32) into LDS, xor-swizzled ----
#if defined(__AMDGCN__) && defined(__gfx1250__)
    {
        // CDNA5 async global->LDS copy: no VGPR staging, tracked by ASYNCcnt.
        const unsigned lds_base = (unsigned)(size_t)(void*)smem;
        const float* mbase = msg + (size_t)n * (KE * 384);
        for (int i = tid; i < KE * 96; i += THREADS) {          // 12 iters/thread
            int k = i / 96;
            int c = (i % 96) << 2;
            int sw = (k & 15) << 2;
            unsigned lds  = lds_base + (unsigned)((k * XSTRIDE + (c ^ sw)) << 2);
            unsigned goff = (unsigned)i << 4;                   // bytes
            asm volatile("global_load_async_to_lds_b128 %0, %1, %2"
                         :: "v"(lds), "v"(goff), "s"(mbase) : "memory");
        }
        const float* ebase = e_repr + (size_t)n * (KE * EDIM);
        for (int i = tid; i < KE * 32; i += THREADS) {          // 4 iters/thread
            int k = i / 32;
            int c = 384 + ((i % 32) << 2);
            int sw = (k & 15) << 2;
            unsigned lds  = lds_base + (unsigned)((k * XSTRIDE + (c ^ sw)) << 2);
            unsigned goff = (unsigned)i << 4;                   // bytes
            asm volatile("global_load_async_to_lds_b128 %0, %1, %2"
                         :: "v"(lds), "v"(goff), "s"(ebase) : "memory");
        }
        asm volatile("s_wait_asynccnt 0x0" ::: "memory");
    }
#else
    {
        const float4* m4 = (const float4*)(msg + (size_t)n * KE * 384);
        for (int i = tid; i < KE * 96; i += THREADS) {
            int k = i / 96;
            int c = (i % 96) << 2;
            float4 v = m4[i];
            int sw = (k & 15) << 2;
            *(float4*)&smem[k * XSTRIDE + (c ^ sw)] = v;
        }
        const float4* e4 = (const float4*)(e_repr + (size_t)n * KE * EDIM);
        for (int i = tid; i < KE * 32; i += THREADS) {
            int k = i / 32;
            int c = 384 + ((i % 32) << 2);
            float4 v = e4[i];
            int sw = (k & 15) << 2;
            *(float4*)&smem[k * XSTRIDE + (c ^ sw)] = v;
        }
    }
#endif
    __syncthreads();

    // ---- stage 1: h = relu(x @ W1 + b1) via V_WMMA_F32_16X16X4_F32 ----
    // wave w computes columns [16w, 16w+16); acc0 = rows 0..15, acc1 = rows 16..31
    const int col = (wave << 4) + l15;
    v8f acc0 = {};
    v8f acc1 = {};
    const float* Wc  = W1 + col;        // column of W1 (row-major 512x128)
    const int    sw0 = l15 << 2;
    #pragma unroll 4
    for (int kk = 0; kk < FDIM / 4; ++kk) {
        const int kb = (kk << 2) + (half << 1);   // K base for this lane half
        // A fragments: x[row][kb], x[row][kb+1]  (one b64-class LDS read each)
        v2f a0 = *(const v2f*)&smem[ l15       * XSTRIDE + (kb ^ sw0)];
        v2f a1 = *(const v2f*)&smem[(l15 + 16) * XSTRIDE + (kb ^ sw0)];
        // B fragment: W1[kb][col], W1[kb+1][col]  (L2-resident weights)
        v2f b;
        b.x = Wc[(size_t)kb * EDIM];
        b.y = Wc[(size_t)(kb + 1) * EDIM];
        acc0 = __builtin_amdgcn_wmma_f32_16x16x4_f32(false, a0, false, b,
                                                     (short)0, acc0, false, false);
        acc1 = __builtin_amdgcn_wmma_f32_16x16x4_f32(false, a1, false, b,
                                                     (short)0, acc1, false, false);
    }

    // ---- epilogue: bias + relu + alpha-weighted k-reduction ----
    // C layout: acc0[j] = h[j + 8*half][col], acc1[j] = h[16 + j + 8*half][col]
    const float bv = b1[col];
    float s = 0.f;
    #pragma unroll
    for (int j = 0; j < 8; ++j) {
        const int m0 = j + (half << 3);
        float h0 = acc0[j] + bv; h0 = h0 > 0.f ? h0 : 0.f;
        float h1 = acc1[j] + bv; h1 = h1 > 0.f ? h1 : 0.f;
        s = fmaf(smem[SMEM_ALPHA + m0],      h0, s);
        s = fmaf(smem[SMEM_ALPHA + m0 + 16], h1, s);
    }
    s += __shfl_xor(s, 16, 32);           // combine lane pair (L, L+16): full column sum

    __syncthreads();                      // x no longer needed: reuse LDS
    if (half == 0) smem[col] = s;         // nei  -> smem[0..128)
    if (tid < EDIM)                       // ce   -> smem[128..256)
        smem[EDIM + tid] = curr_emb[(size_t)n * KE * EDIM + tid];
    __syncthreads();

    // ---- stage 2: out = relu([ce, nei] @ W2 + b2)  (1x256 @ 256x128, VALU) ----
    if (tid < EDIM) {
        float acc = b2[tid];
        #pragma unroll 8
        for (int g = 0; g < EDIM; ++g)                 // ce part: W2 rows [0,128)
            acc = fmaf(smem[EDIM + g], W2[(size_t)g * EDIM + tid], acc);
        #pragma unroll 8
        for (int g = 0; g < EDIM; ++g)                 // nei part: W2 rows [128,256)
            acc = fmaf(smem[g], W2[(size_t)(EDIM + g) * EDIM + tid], acc);
        out[(size_t)n * EDIM + tid] = acc > 0.f ? acc : 0.f;
    }
}

extern "C" void kernel_launch(void* const* d_in, const int* in_sizes, int n_in,
                              void* d_out, int out_size, void* d_ws, size_t ws_size,
                              hipStream_t stream) {
    const float* msg      = (const float*)d_in[0];
    const float* e_repr   = (const float*)d_in[1];
    const float* alpha    = (const float*)d_in[2];
    const float* curr_emb = (const float*)d_in[3];
    const float* W1       = (const float*)d_in[4];
    const float* b1       = (const float*)d_in[5];
    const float* W2       = (const float*)d_in[6];
    const float* b2       = (const float*)d_in[7];
    float* out            = (float*)d_out;

    const int N = in_sizes[0] / (KE * 3 * EDIM);      // msg is (N, 32, 384)
    const size_t smem_bytes = (size_t)SMEM_FLOATS * sizeof(float);

    edge_agg_kernel<<<N, THREADS, smem_bytes, stream>>>(
        msg, e_repr, alpha, curr_emb, W1, b1, W2, b2, out);
}